// GINE_L_8564164788538
// MI455X (gfx1250) — compile-verified
//
#include <hip/hip_runtime.h>

// ---------------------------------------------------------------------------
// GINE 3-layer GNN for MI455X (gfx1250, wave32).
//  - Edge scatter fused with rank-1 edge embedding + ReLU (atomic f32 adds, L2
//    resident: node features <= 25.6MB << 192MB L2).
//  - Node MLPs + head MLP via V_WMMA_F32_16X16X4_F32, one wave per 16-row tile.
//  - BN stats accumulated in-kernel via lane-pair shuffle + global atomics.
// ---------------------------------------------------------------------------

typedef __attribute__((ext_vector_type(2))) float v2f;
typedef __attribute__((ext_vector_type(8))) float v8f;

// One 16x16 f32 output tile: C += A(16xK, row-major lda) * B(KxN tile, row-major ldb).
// VGPR layouts per CDNA5 ISA 7.12.2:
//   A 16x4 f32:  lanes 0-15 rows 0-15 {K=k,k+1}; lanes 16-31 rows 0-15 {K=k+2,k+3}
//   B 4x16 f32:  mirrored striping (row k in vgpr0 lanes 0-15, k+2 lanes 16-31, ...)
//   C 16x16 f32: vgpr r -> row r (lanes 0-15, N=lane) / row r+8 (lanes 16-31, N=lane-16)
__device__ __forceinline__ v8f gemm16_tile(const float* A, int lda,
                                           const float* B, int ldb,
                                           int K, int lane) {
  const int half = lane >> 4;   // 0 or 1
  const int l15  = lane & 15;
  v8f c = {};
  for (int k = 0; k < K; k += 4) {
    const int ka = k + 2 * half;
    v2f a, b;
    a.x = A[l15 * lda + ka];
    a.y = A[l15 * lda + ka + 1];
    b.x = B[ka * ldb + l15];
    b.y = B[(ka + 1) * ldb + l15];
    c = __builtin_amdgcn_wmma_f32_16x16x4_f32(false, a, false, b,
                                              (short)0, c, false, false);
  }
  return c;
}

// ---------------------------- elementwise helpers ---------------------------

__global__ void copy_kernel(float* __restrict__ dst, const float* __restrict__ src,
                            long long n) {
  long long i = (long long)blockIdx.x * blockDim.x + threadIdx.x;
  const long long stride = (long long)gridDim.x * blockDim.x;
  for (; i < n; i += stride) dst[i] = src[i];
}

__global__ void zero_kernel(float* __restrict__ dst, long long n) {
  long long i = (long long)blockIdx.x * blockDim.x + threadIdx.x;
  const long long stride = (long long)gridDim.x * blockDim.x;
  for (; i < n; i += stride) dst[i] = 0.0f;
}

// ----------------------- fused edge message + scatter -----------------------
// acc[dst] += relu(x[src] + edge_attr[e] * we_w + we_b), 4 features per thread.
__global__ void edge_scatter_kernel(const float* __restrict__ x,
                                    const int* __restrict__ src,
                                    const int* __restrict__ dst,
                                    const float* __restrict__ ea,
                                    const float* __restrict__ wew,
                                    const float* __restrict__ web,
                                    float* __restrict__ acc,
                                    long long E, int din, int gshift) {
  const int groups = din >> 2;                 // float4 groups per edge
  const long long total = E * (long long)groups;
  long long i = (long long)blockIdx.x * blockDim.x + threadIdx.x;
  const long long stride = (long long)gridDim.x * blockDim.x;
  for (; i < total; i += stride) {
    const long long e = i >> gshift;           // gshift = log2(groups)
    const int c = (int)(i & (groups - 1)) << 2;
    const int s = src[e];
    const int d = dst[e];
    const float a = ea[e];
    const float4 xv = *reinterpret_cast<const float4*>(x + (long long)s * din + c);
    float* ap = acc + (long long)d * din + c;
    float m;
    m = xv.x + a * wew[c + 0] + web[c + 0]; unsafeAtomicAdd(ap + 0, m > 0.f ? m : 0.f);
    m = xv.y + a * wew[c + 1] + web[c + 1]; unsafeAtomicAdd(ap + 1, m > 0.f ? m : 0.f);
    m = xv.z + a * wew[c + 2] + web[c + 2]; unsafeAtomicAdd(ap + 2, m > 0.f ? m : 0.f);
    m = xv.w + a * wew[c + 3] + web[c + 3]; unsafeAtomicAdd(ap + 3, m > 0.f ? m : 0.f);
  }
}

// --------------------- per-node 2-layer MLP + BN stats ----------------------
// zraw = relu( relu(in@w1 + b1) @ w2 + b2 ); stats[0..dh)=sum, stats[dh..2dh)=sum^2.
// One wave (32 threads) per 16-node tile. din, dh in {32,64,128}.
__global__ void __launch_bounds__(32)
mlp_bn_kernel(const float* __restrict__ in,
              const float* __restrict__ w1, const float* __restrict__ b1,
              const float* __restrict__ w2, const float* __restrict__ b2,
              float* __restrict__ zraw, float* __restrict__ stats,
              long long N, int din, int dh) {
  __shared__ float hid[16 * 128];
  const int lane = threadIdx.x;
  const int half = lane >> 4;
  const int l15  = lane & 15;
  const long long row0 = (long long)blockIdx.x * 16;

  // Clamp A row for safety if N not a multiple of 16 (it is: 50000 = 3125*16).
  long long arow = row0 + l15;
  if (arow >= N) arow = N - 1;
  const float* A = in + (arow - l15) * din;    // gemm16_tile adds l15*lda back

  // GEMM 1: hidden = relu(A @ w1 + b1)
  for (int nt = 0; nt < dh; nt += 16) {
    v8f c = gemm16_tile(A, din, w1 + nt, dh, din, lane);
    const int col = nt + l15;
    const float bv = b1[col];
#pragma unroll
    for (int r = 0; r < 8; ++r) {
      float v = c[r] + bv;
      hid[(r + 8 * half) * dh + col] = v > 0.f ? v : 0.f;
    }
  }
  __syncthreads();

  // GEMM 2: z = relu(hidden @ w2 + b2), accumulate BN sum / sum^2
  for (int nt = 0; nt < dh; nt += 16) {
    v8f c = gemm16_tile(hid, dh, w2 + nt, dh, dh, lane);
    const int col = nt + l15;
    const float bv = b2[col];
    float s = 0.f, s2 = 0.f;
#pragma unroll
    for (int r = 0; r < 8; ++r) {
      const long long row = row0 + r + 8 * half;
      float v = c[r] + bv;
      v = v > 0.f ? v : 0.f;
      if (row < N) {
        zraw[row * dh + col] = v;
        s += v;
        s2 += v * v;
      }
    }
    // lanes L and L^16 hold the same output column -> pairwise combine
    s  += __shfl_xor(s, 16);
    s2 += __shfl_xor(s2, 16);
    if (lane < 16) {
      unsafeAtomicAdd(&stats[col], s);
      unsafeAtomicAdd(&stats[dh + col], s2);
    }
  }
}

// ----------------------------- batchnorm tail -------------------------------

__global__ void bn_finalize_kernel(const float* __restrict__ stats,
                                   const float* __restrict__ g,
                                   const float* __restrict__ b,
                                   float* __restrict__ scsh,
                                   float invN, int dh) {
  const int i = threadIdx.x;
  if (i < dh) {
    const float mu  = stats[i] * invN;
    const float var = stats[dh + i] * invN - mu * mu;
    const float sc  = g[i] * rsqrtf(var + 1e-5f);
    scsh[i]      = sc;
    scsh[dh + i] = b[i] - mu * sc;
  }
}

__global__ void bn_apply_kernel(const float* __restrict__ zraw,
                                const float* __restrict__ scsh,
                                float* __restrict__ z,
                                long long n, int dhmask, int dh) {
  long long i = (long long)blockIdx.x * blockDim.x + threadIdx.x;
  const long long stride = (long long)gridDim.x * blockDim.x;
  for (; i < n; i += stride) {
    const int col = (int)(i & dhmask);
    z[i] = zraw[i] * scsh[col] + scsh[dh + col];
  }
}

// ------------------------------- pooling ------------------------------------

__global__ void pool_kernel(const float* __restrict__ z,
                            const int* __restrict__ batch,
                            float* __restrict__ pooled, long long n) {  // n = N*32
  long long i = (long long)blockIdx.x * blockDim.x + threadIdx.x;
  const long long stride = (long long)gridDim.x * blockDim.x;
  for (; i < n; i += stride) {
    const long long node = i >> 5;
    const int f = (int)(i & 31);
    unsafeAtomicAdd(&pooled[(long long)batch[node] * 32 + f], z[i]);
  }
}

// --------------------------------- head -------------------------------------
// out = relu(relu(pooled@fc1+b1)@fc2+b2) @ fc3 + b3 ; 16 rows per wave.
__global__ void __launch_bounds__(32)
head_kernel(const float* __restrict__ pooled,
            const float* __restrict__ fc1w, const float* __restrict__ fc1b,
            const float* __restrict__ fc2w, const float* __restrict__ fc2b,
            const float* __restrict__ fc3w, const float* __restrict__ fc3b,
            float* __restrict__ out, int G) {
  __shared__ float h1[16 * 128];
  __shared__ float h2[16 * 64];
  const int lane = threadIdx.x;
  const int half = lane >> 4;
  const int l15  = lane & 15;
  const int row0 = blockIdx.x * 16;
  const float* A = pooled + (long long)row0 * 32;

  for (int nt = 0; nt < 128; nt += 16) {           // fc1: 32 -> 128
    v8f c = gemm16_tile(A, 32, fc1w + nt, 128, 32, lane);
    const int col = nt + l15;
    const float bv = fc1b[col];
#pragma unroll
    for (int r = 0; r < 8; ++r) {
      float v = c[r] + bv;
      h1[(r + 8 * half) * 128 + col] = v > 0.f ? v : 0.f;
    }
  }
  __syncthreads();

  for (int nt = 0; nt < 64; nt += 16) {            // fc2: 128 -> 64
    v8f c = gemm16_tile(h1, 128, fc2w + nt, 64, 128, lane);
    const int col = nt + l15;
    const float bv = fc2b[col];
#pragma unroll
    for (int r = 0; r < 8; ++r) {
      float v = c[r] + bv;
      h2[(r + 8 * half) * 64 + col] = v > 0.f ? v : 0.f;
    }
  }
  __syncthreads();

  // fc3: 64 -> 2, scalar (16 rows x 2 cols across 32 lanes)
  const int row = lane >> 1;
  const int col = lane & 1;
  float acc = fc3b[col];
  for (int k = 0; k < 64; ++k) acc += h2[row * 64 + k] * fc3w[k * 2 + col];
  if (row0 + row < G) out[(long long)(row0 + row) * 2 + col] = acc;
}

// ------------------------------ orchestration -------------------------------
// Input order (setup_inputs insertion order, nested params flattened in
// insertion order): x, edge_index, edge_attr, batch,
//   [gin{0,1,2}: we_w, we_b, w1, b1, w2, b2, bn_g, bn_b] x3,
//   fc1_w, fc1_b, fc2_w, fc2_b, fc3_w, fc3_b.

extern "C" void kernel_launch(void* const* d_in, const int* in_sizes, int n_in,
                              void* d_out, int out_size, void* d_ws, size_t ws_size,
                              hipStream_t stream) {
  const float* x    = (const float*)d_in[0];
  const int*  ei    = (const int*)d_in[1];
  const float* ea   = (const float*)d_in[2];
  const int*  batch = (const int*)d_in[3];

  const long long N = in_sizes[0] / 32;
  const long long E = in_sizes[1] / 2;
  const int G = out_size / 2;

  const int* srcIdx = ei;
  const int* dstIdx = ei + E;

  const float* P[30];
  for (int i = 0; i < 30 && (4 + i) < n_in; ++i) P[i] = (const float*)d_in[4 + i];

  float* ws     = (float*)d_ws;
  float* bufA   = ws;                  // z (post-BN) [N,128]
  float* bufB   = bufA + N * 128;      // acc = x + scatter [N,128]
  float* bufC   = bufB + N * 128;      // z_raw (pre-BN)    [N,128]
  float* stats  = bufC + N * 128;      // [2*128]
  float* scsh   = stats + 256;         // [2*128]
  float* pooled = scsh + 256;          // [G*32]
  (void)ws_size;

  const int dins[3] = {32, 128, 64};
  const int dhs[3]  = {128, 64, 32};

  const float* zprev = x;
  for (int l = 0; l < 3; ++l) {
    const int din = dins[l];
    const int dh  = dhs[l];
    const float* wew = P[l * 8 + 0];
    const float* web = P[l * 8 + 1];
    const float* w1  = P[l * 8 + 2];
    const float* b1  = P[l * 8 + 3];
    const float* w2  = P[l * 8 + 4];
    const float* b2  = P[l * 8 + 5];
    const float* bng = P[l * 8 + 6];
    const float* bnb = P[l * 8 + 7];

    // residual: acc starts as a copy of the layer input
    copy_kernel<<<2048, 256, 0, stream>>>(bufB, zprev, N * (long long)din);

    const int gshift = (din == 32) ? 3 : (din == 128) ? 5 : 4;  // log2(din/4)
    const long long totalE = E * (long long)(din >> 2);
    long long be = (totalE + 255) / 256;
    if (be > 131072) be = 131072;
    edge_scatter_kernel<<<(int)be, 256, 0, stream>>>(zprev, srcIdx, dstIdx, ea,
                                                     wew, web, bufB, E, din, gshift);

    zero_kernel<<<1, 256, 0, stream>>>(stats, 2LL * dh);
    mlp_bn_kernel<<<(int)((N + 15) / 16), 32, 0, stream>>>(bufB, w1, b1, w2, b2,
                                                           bufC, stats, N, din, dh);
    bn_finalize_kernel<<<1, 128, 0, stream>>>(stats, bng, bnb, scsh,
                                              1.0f / (float)N, dh);
    bn_apply_kernel<<<2048, 256, 0, stream>>>(bufC, scsh, bufA,
                                              N * (long long)dh, dh - 1, dh);
    zprev = bufA;
  }

  zero_kernel<<<32, 256, 0, stream>>>(pooled, (long long)G * 32);
  pool_kernel<<<2048, 256, 0, stream>>>(bufA, batch, pooled, N * 32LL);
  head_kernel<<<(G + 15) / 16, 32, 0, stream>>>(pooled, P[24], P[25], P[26], P[27],
                                                P[28], P[29], (float*)d_out, G);
}